// TrilinearScorer_72224170049657
// MI455X (gfx1250) — compile-verified
//
#include <hip/hip_runtime.h>
#include <stdint.h>

typedef __attribute__((ext_vector_type(2))) float v2f;
typedef __attribute__((ext_vector_type(8))) float v8f;

#define BM 64
#define BN 128
#define KC 16
#define NTHREADS 256
#define KSTRIDE 20   // padded row stride (floats): rows 80B (16B-aligned), 16 lanes @ stride 20 -> 16 distinct banks

// ---- CDNA5 async global->LDS copies (ASYNCcnt-tracked), CDNA5 ISA
__device__ __forceinline__ unsigned lds_off(const void* p) {
  return (unsigned)(uintptr_t)p;  // low 32 bits of flat LDS address = LDS byte offset
}
__device__ __forceinline__ void async_b32(unsigned lds, const float* base, unsigned boff) {
  asm volatile("global_load_async_to_lds_b32 %0, %1, %2"
               :: "v"(lds), "v"(boff), "s"(base) : "memory");
}
__device__ __forceinline__ void async_b128(unsigned lds, const float* base, unsigned boff) {
  asm volatile("global_load_async_to_lds_b128 %0, %1, %2"
               :: "v"(lds), "v"(boff), "s"(base) : "memory");
}
__device__ __forceinline__ void wait_async() {
  asm volatile("s_wait_asynccnt 0x0" ::: "memory");
}

// One KC=16 chunk of V_WMMA_F32_16X16X4_F32 for this wave's 32x32 patch (2x2 tiles).
// As[m][k] holds Aeff[m][k]; Bs[n][k] holds Beff[k][n] (transposed: symmetric b64 frag loads).
__device__ __forceinline__ void chunk_mma(v8f (&acc)[2][2],
                                          const float (&As)[BM][KSTRIDE],
                                          const float (&Bs)[BN][KSTRIDE],
                                          int wm, int wn, int half, int l16) {
  #pragma unroll
  for (int k4 = 0; k4 < KC; k4 += 4) {
    v2f aF[2], bF[2];
    #pragma unroll
    for (int tm = 0; tm < 2; ++tm) {
      // fp32 16x4 A frag: v0 <- K=2*half, v1 <- K=2*half+1, M = lane&15
      int m = wm + tm * 16 + l16;
      aF[tm].x = As[m][k4 + 2 * half];
      aF[tm].y = As[m][k4 + 2 * half + 1];
    }
    #pragma unroll
    for (int tn = 0; tn < 2; ++tn) {
      // fp32 4x16 B frag: row K = v + 2*half, N = lane&15 (Bs stored N-major)
      int n = wn + tn * 16 + l16;
      bF[tn].x = Bs[n][k4 + 2 * half];
      bF[tn].y = Bs[n][k4 + 2 * half + 1];
    }
    #pragma unroll
    for (int tm = 0; tm < 2; ++tm)
      #pragma unroll
      for (int tn = 0; tn < 2; ++tn)
        acc[tm][tn] = __builtin_amdgcn_wmma_f32_16x16x4_f32(
            false, aF[tm], false, bF[tn], (short)0, acc[tm][tn], false, false);
  }
}

// Body specialized for interior (EDGE=false: no guards at all) vs edge blocks.
// K = 257 fixed: 16 guard-free chunks + 1 tail chunk (append-ones / col 256 / zero pad).
template <bool A_ONES, bool B_ONES, bool B_TRANS, bool EDGE>
__device__ __forceinline__ void gemm_body(
    const float* __restrict__ A, const float* __restrict__ B, float* __restrict__ C,
    int N, long long lda, long long ldb, long long ldc,
    int m0, int n0, int t, int wave, int half, int l16,
    float (&As)[BM][KSTRIDE], float (&Bs)[BN][KSTRIDE])
{
  constexpr bool A_VEC4 = A_ONES;   // stages 1/2: lda == 256 -> 16B-aligned rows
  constexpr bool B_VEC4 = B_TRANS;  // stage 3:   ldb == 256, k-contiguous rows

  const int wm = (wave >> 2) * 32;
  const int wn = (wave & 3) * 32;

  // ---- precompute loop-invariant LDS destinations + global byte offsets
  unsigned ldsA[A_VEC4 ? 1 : 4], offA[A_VEC4 ? 1 : 4];
  if (A_VEC4) {
    int m = t >> 2, kk = (t & 3) * 4;            // 64 rows x 4 b128 groups
    ldsA[0] = lds_off(&As[m][kk]);
    offA[0] = (unsigned)(((long long)(m0 + m) * lda + kk) * 4);
  } else {
    #pragma unroll
    for (int e = 0; e < 4; ++e) {
      int p = t + e * NTHREADS, m = p >> 4, kk = p & (KC - 1);
      ldsA[e] = lds_off(&As[m][kk]);
      offA[e] = (unsigned)(((long long)(m0 + m) * lda + kk) * 4);
    }
  }
  unsigned ldsB[B_VEC4 ? 2 : 8], offB[B_VEC4 ? 2 : 8];
  bool vB[B_VEC4 ? 2 : 8];
  if (B_VEC4) {
    #pragma unroll
    for (int e = 0; e < 2; ++e) {
      int p = t + e * NTHREADS, nn = p >> 2, kk = (p & 3) * 4;  // 128 rows x 4 b128 groups
      ldsB[e] = lds_off(&Bs[nn][kk]);
      offB[e] = (unsigned)(((long long)(n0 + nn) * ldb + kk) * 4);
      vB[e] = (n0 + nn) < N;
    }
  } else {
    #pragma unroll
    for (int e = 0; e < 8; ++e) {
      int p = t + e * NTHREADS, kk = p >> 7, nn = p & (BN - 1);
      ldsB[e] = lds_off(&Bs[nn][kk]);
      offB[e] = (unsigned)(((long long)kk * ldb + n0 + nn) * 4);
      vB[e] = (n0 + nn) < N;
    }
  }
  const unsigned stepA = KC * 4;                                   // k += 16 along a row
  const unsigned stepB = B_TRANS ? (KC * 4) : (unsigned)(KC * ldb * 4);

  // Edge blocks: zero columns with n >= N once; async loads never touch them.
  if (EDGE) {
    float* bflat = &Bs[0][0];
    for (int p = t; p < BN * KSTRIDE; p += NTHREADS) bflat[p] = 0.0f;
  }

  v8f acc[2][2] = {};

  // ---- 16 guard-free chunks: k = 0..255 is real data for A and B in every stage
  for (int kb = 0; kb < 16; ++kb) {
    if (A_VEC4) {
      async_b128(ldsA[0], A, offA[0]);
      offA[0] += stepA;
    } else {
      #pragma unroll
      for (int e = 0; e < 4; ++e) { async_b32(ldsA[e], A, offA[e]); offA[e] += stepA; }
    }
    if (B_VEC4) {
      #pragma unroll
      for (int e = 0; e < 2; ++e) {
        if (!EDGE || vB[e]) async_b128(ldsB[e], B, offB[e]);
        offB[e] += stepB;
      }
    } else {
      #pragma unroll
      for (int e = 0; e < 8; ++e) {
        if (!EDGE || vB[e]) async_b32(ldsB[e], B, offB[e]);
        offB[e] += stepB;
      }
    }
    wait_async();
    __syncthreads();
    chunk_mma(acc, As, Bs, wm, wn, half, l16);
    __syncthreads();
  }

  // ---- tail chunk: k = 256..271
  {
    #pragma unroll
    for (int e = 0; e < 4; ++e) {
      int p = t + e * NTHREADS, m = p >> 4, kk = p & (KC - 1);
      float v = 0.0f;
      if (kk == 0) v = A_ONES ? 1.0f : A[(long long)(m0 + m) * lda + 256];
      As[m][kk] = v;
    }
    #pragma unroll
    for (int e = 0; e < 8; ++e) {
      int p = t + e * NTHREADS, nn = p >> 4, kk = p & (KC - 1);
      int n = n0 + nn;
      float v = 0.0f;
      if (kk == 0 && (!EDGE || n < N))
        v = B_ONES ? 1.0f
                   : (B_TRANS ? B[(long long)n * ldb + 256]
                              : B[(long long)256 * ldb + n]);
      Bs[nn][kk] = v;
    }
    __syncthreads();
    chunk_mma(acc, As, Bs, wm, wn, half, l16);
  }

  // ---- store: C/D layout VGPR r -> M = r + 8*half, N = lane&15
  #pragma unroll
  for (int tm = 0; tm < 2; ++tm) {
    #pragma unroll
    for (int tn = 0; tn < 2; ++tn) {
      int n = n0 + wn + tn * 16 + l16;
      if (!EDGE || n < N) {
        #pragma unroll
        for (int r = 0; r < 8; ++r) {
          int m = m0 + wm + tm * 16 + 8 * half + r;
          C[(long long)m * ldc + n] = acc[tm][tn][r];
        }
      }
    }
  }
}

template <bool A_ONES, bool B_ONES, bool B_TRANS>
__global__ __launch_bounds__(NTHREADS) void gemm_wmma_f32(
    const float* __restrict__ A, const float* __restrict__ B,
    float* __restrict__ C,
    int N, long long lda, long long ldb, long long ldc,
    int A_gdiv, long long A_gstride,
    int B_gdiv, long long B_gstride,
    long long C_gstride)
{
  __shared__ float As[BM][KSTRIDE];
  __shared__ float Bs[BN][KSTRIDE];

  const int t    = threadIdx.x;
  const int lane = t & 31;       // wave32
  const int wave = t >> 5;
  const int half = lane >> 4;
  const int l16  = lane & 15;

  const int g = blockIdx.z;
  A += (long long)(g / A_gdiv) * A_gstride;
  B += (long long)(g / B_gdiv) * B_gstride;
  C += (long long)g * C_gstride;

  const int m0 = blockIdx.y * BM;
  const int n0 = blockIdx.x * BN;

  if (n0 + BN <= N)  // uniform per block
    gemm_body<A_ONES, B_ONES, B_TRANS, false>(A, B, C, N, lda, ldb, ldc,
                                              m0, n0, t, wave, half, l16, As, Bs);
  else
    gemm_body<A_ONES, B_ONES, B_TRANS, true>(A, B, C, N, lda, ldb, ldc,
                                             m0, n0, t, wave, half, l16, As, Bs);
}

// Problem constants (B=2, L=256, D=256, A=257)
#define NB 2
#define LL 256
#define DD 256
#define AA 257          // D+1
#define NI 512          // NB*LL  (n,i) pairs
#define NBC 66049       // 257*257
#define T2ROW 65792     // 256*257

extern "C" void kernel_launch(void* const* d_in, const int* in_sizes, int n_in,
                              void* d_out, int out_size, void* d_ws, size_t ws_size,
                              hipStream_t stream) {
  (void)in_sizes; (void)n_in; (void)out_size; (void)ws_size;
  const float* l1 = (const float*)d_in[0];   // [2,256,256]
  const float* l2 = (const float*)d_in[1];   // [2,256,256]
  const float* l3 = (const float*)d_in[2];   // [2,256,256]
  const float* W  = (const float*)d_in[3];   // [257,257,257]
  float* out = (float*)d_out;                // [2,256,256,256]

  float* T1 = (float*)d_ws;                  // [512][66049]     (135 MB)
  float* T2 = T1 + (size_t)NI * NBC;         // [512][256][257]  (135 MB)

  dim3 blk(NTHREADS);

  // Stage 1: T1[(n,i)][b*257+c] = l1aug @ Wflat      M=512, K=257, N=66049
  dim3 g1((NBC + BN - 1) / BN, NI / BM, 1);
  gemm_wmma_f32<true, false, false><<<g1, blk, 0, stream>>>(
      l1, W, T1,
      /*N*/ NBC, /*lda*/ DD, /*ldb*/ NBC, /*ldc*/ NBC,
      /*A_gdiv*/ 1, /*A_gstride*/ 0, /*B_gdiv*/ 1, /*B_gstride*/ 0,
      /*C_gstride*/ 0);

  // Stage 2: T2_g[j][c] = l2aug[n] @ T1_g            512 x (M=256, K=257, N=257)
  dim3 g2((AA + BN - 1) / BN, LL / BM, NI);
  gemm_wmma_f32<true, false, false><<<g2, blk, 0, stream>>>(
      l2, T1, T2,
      /*N*/ AA, /*lda*/ DD, /*ldb*/ AA, /*ldc*/ AA,
      /*A_gdiv*/ LL, /*A_gstride*/ (long long)LL * DD,
      /*B_gdiv*/ 1,  /*B_gstride*/ NBC,
      /*C_gstride*/ T2ROW);

  // Stage 3: OUT_g[j][k] = T2_g @ l3aug[n]^T         512 x (M=256, K=257, N=256)
  dim3 g3((LL + BN - 1) / BN, LL / BM, NI);
  gemm_wmma_f32<false, true, true><<<g3, blk, 0, stream>>>(
      T2, l3, out,
      /*N*/ LL, /*lda*/ AA, /*ldb*/ DD, /*ldc*/ LL,
      /*A_gdiv*/ 1,  /*A_gstride*/ T2ROW,
      /*B_gdiv*/ LL, /*B_gstride*/ (long long)LL * DD,
      /*C_gstride*/ (long long)LL * LL);
}